// SpatialConsistencyLoss_66503273611692
// MI455X (gfx1250) — compile-verified
//
#include <hip/hip_runtime.h>
#include <hip/hip_bf16.h>
#include <math.h>

// ---------------------------------------------------------------------------
// SpatialConsistencyLoss for MI455X (gfx1250, wave32)
//   Gram tiles via V_WMMA_F32_16X16X4_F32 (K=3 padded to 4), top-8 selection
//   in registers, deterministic two-stage reduction.
// ---------------------------------------------------------------------------

typedef __attribute__((ext_vector_type(2))) float v2f;
typedef __attribute__((ext_vector_type(8))) float v8f;

#define ROWS_PER_BLOCK 128
#define CHUNK          1024
#define KNN            8

__global__ __launch_bounds__(ROWS_PER_BLOCK) void scl_knn_dist_kernel(
    const float* __restrict__ xyz,   // (4, 3, npts) contiguous
    float* __restrict__ dist,        // (4, npts)
    int npts)
{
    const int b    = blockIdx.y;
    const int tid  = threadIdx.x;
    const int lane = tid & 31;
    const int wave = tid >> 5;
    const int half = lane >> 4;      // 0: K={0,1}, 1: K={2,3}
    const int l16  = lane & 15;
    const int row0 = blockIdx.x * ROWS_PER_BLOCK;

    const float* __restrict__ X = xyz + (size_t)b * 3 * npts;
    const float* __restrict__ Y = X + npts;
    const float* __restrict__ Z = Y + npts;

    __shared__ float rowc[ROWS_PER_BLOCK][4];   // x, y, z, |p|^2
    __shared__ float colc[CHUNK][4];            // x, y, z, |p|^2
    __shared__ float d2t[ROWS_PER_BLOCK][17];   // padded to dodge bank conflicts

    // ---- stage this block's row points -----------------------------------
    {
        int i  = row0 + tid;
        int ic = (i < npts) ? i : (npts - 1);
        float x = X[ic], y = Y[ic], z = Z[ic];
        rowc[tid][0] = x; rowc[tid][1] = y; rowc[tid][2] = z;
        rowc[tid][3] = x * x + y * y + z * z;
    }
    __syncthreads();

    const int   myRow = row0 + tid;
    const float px = rowc[tid][0], py = rowc[tid][1], pz = rowc[tid][2];

    // sorted ascending top-8 candidate list (registers)
    float bd[KNN];
    int   bj[KNN];
#pragma unroll
    for (int k = 0; k < KNN; ++k) { bd[k] = 3.0e38f; bj[k] = 0; }

    // A-matrix operands for this wave's two 16-row sub-tiles (loop invariant)
    // A is 16x4: lane m=l16; VGPR0 = K(2*half), VGPR1 = K(2*half+1); K3 = 0.
    v2f aT[2];
    float rsq[2][8];                  // row |p|^2 per (t, v) — loop invariant
#pragma unroll
    for (int t = 0; t < 2; ++t) {
        int m = wave * 32 + t * 16 + l16;
        aT[t][0] = (half == 0) ? rowc[m][0] : rowc[m][2];
        aT[t][1] = (half == 0) ? rowc[m][1] : 0.0f;
#pragma unroll
        for (int v = 0; v < 8; ++v)
            rsq[t][v] = rowc[wave * 32 + t * 16 + half * 8 + v][3];
    }

    // ---- sweep all columns in LDS chunks ---------------------------------
    for (int cbase = 0; cbase < npts; cbase += CHUNK) {
        const int clen = min(CHUNK, npts - cbase);

        __syncthreads();
        for (int c = tid; c < CHUNK; c += ROWS_PER_BLOCK) {
            int j  = cbase + c;
            int jc = (j < npts) ? j : (npts - 1);
            float x = X[jc], y = Y[jc], z = Z[jc];
            colc[c][0] = x; colc[c][1] = y; colc[c][2] = z;
            colc[c][3] = x * x + y * y + z * z;
        }
        __syncthreads();

        const int ntiles = (clen + 15) >> 4;
        for (int jt = 0; jt < ntiles; ++jt) {
            const int n = jt * 16 + l16;          // column within chunk
            // B is 4x16: lane n=l16; VGPR0 = K(2*half), VGPR1 = K(2*half+1)
            v2f bv;
            bv[0] = (half == 0) ? colc[n][0] : colc[n][2];
            bv[1] = (half == 0) ? colc[n][1] : 0.0f;
            const float sqc = colc[n][3];

#pragma unroll
            for (int t = 0; t < 2; ++t) {
                v8f acc = {};
                // D = A x B + 0 : full-fp32 Gram tile, K padded 3 -> 4
                acc = __builtin_amdgcn_wmma_f32_16x16x4_f32(
                    /*neg_a=*/false, aT[t], /*neg_b=*/false, bv,
                    /*c_mod=*/(short)0, acc, /*reuse_a=*/false, /*reuse_b=*/false);
                // C/D layout: lane holds col N=l16, rows M = v + 8*half
#pragma unroll
                for (int v = 0; v < 8; ++v) {
                    int   mloc = wave * 32 + t * 16 + half * 8 + v;
                    float d2   = rsq[t][v] + sqc - 2.0f * acc[v];
                    d2t[mloc][l16] = fmaxf(d2, 0.0f);
                }
            }
            __syncthreads();

            // each thread scans its own row's 16 candidates
            const int jbase = cbase + jt * 16;
#pragma unroll
            for (int cc = 0; cc < 16; ++cc) {
                int   j = jbase + cc;
                float d = d2t[tid][cc];
                if ((j < npts) && (j != myRow) && (d < bd[KNN - 1])) {
                    bd[KNN - 1] = d; bj[KNN - 1] = j;
#pragma unroll
                    for (int k = KNN - 1; k > 0; --k) {
                        if (bd[k] < bd[k - 1]) {
                            float td = bd[k]; bd[k] = bd[k - 1]; bd[k - 1] = td;
                            int   tj = bj[k]; bj[k] = bj[k - 1]; bj[k - 1] = tj;
                        }
                    }
                }
            }
            __syncthreads();
        }
    }

    // ---- centroid of 8 NN, distance to it --------------------------------
    if (myRow < npts) {
        float cx = 0.0f, cy = 0.0f, cz = 0.0f;
#pragma unroll
        for (int k = 0; k < KNN; ++k) {
            int j = bj[k];
            cx += X[j]; cy += Y[j]; cz += Z[j];
        }
        const float inv = 1.0f / (float)KNN;
        cx *= inv; cy *= inv; cz *= inv;
        float dx = px - cx, dy = py - cy, dz = pz - cz;
        dist[(size_t)b * npts + myRow] = sqrtf(dx * dx + dy * dy + dz * dz);
    }
}

// ---- per-batch reduction: max(dist), sum(s*dist) -> loss_b ---------------
__global__ __launch_bounds__(256) void scl_reduce_kernel(
    const float* __restrict__ dist,
    const float* __restrict__ score,   // (4, npts)
    float* __restrict__ losses,        // (4,)
    int npts)
{
    const int b   = blockIdx.x;
    const int tid = threadIdx.x;
    __shared__ float red[256];

    const float* __restrict__ d = dist  + (size_t)b * npts;
    const float* __restrict__ s = score + (size_t)b * npts;

    float m = 0.0f;
    for (int i = tid; i < npts; i += 256) m = fmaxf(m, d[i]);
    red[tid] = m;
    __syncthreads();
    for (int off = 128; off > 0; off >>= 1) {
        if (tid < off) red[tid] = fmaxf(red[tid], red[tid + off]);
        __syncthreads();
    }
    const float mx = red[0];
    __syncthreads();

    float acc = 0.0f;
    for (int i = tid; i < npts; i += 256) acc += s[i] * d[i];
    red[tid] = acc;
    __syncthreads();
    for (int off = 128; off > 0; off >>= 1) {
        if (tid < off) red[tid] += red[tid + off];
        __syncthreads();
    }
    if (tid == 0)
        losses[b] = red[0] / ((float)npts * (mx + 1e-6f));
}

__global__ void scl_final_kernel(const float* __restrict__ losses,
                                 float* __restrict__ out)
{
    if (threadIdx.x == 0 && blockIdx.x == 0)
        out[0] = 0.25f * (losses[0] + losses[1] + losses[2] + losses[3]);
}

extern "C" void kernel_launch(void* const* d_in, const int* in_sizes, int n_in,
                              void* d_out, int out_size, void* d_ws, size_t ws_size,
                              hipStream_t stream)
{
    const float* xyz   = (const float*)d_in[0];   // (4, 3, npts, 1) fp32
    const float* score = (const float*)d_in[1];   // (4, 1, npts, 1) fp32
    float*       out   = (float*)d_out;           // scalar fp32

    const int B    = 4;
    const int npts = in_sizes[0] / (B * 3);       // 8000

    float* ws      = (float*)d_ws;
    float* dist    = ws;                          // B * npts floats
    float* losses  = ws + (size_t)B * npts;       // B floats

    dim3 grid1((npts + ROWS_PER_BLOCK - 1) / ROWS_PER_BLOCK, B);
    scl_knn_dist_kernel<<<grid1, ROWS_PER_BLOCK, 0, stream>>>(xyz, dist, npts);

    scl_reduce_kernel<<<B, 256, 0, stream>>>(dist, score, losses, npts);

    scl_final_kernel<<<1, 32, 0, stream>>>(losses, out);
}